// AFE_Layer_55679956025845
// MI455X (gfx1250) — compile-verified
//
#include <hip/hip_runtime.h>
#include <hip/hip_bf16.h>

#define Bn   4
#define Cn   128
#define Hn   96
#define Wn   96
#define OCn  128
#define OFFn 18
#define HWn  (Hn * Wn)
#define NPIX (Bn * HWn)          // 36864
#define AST  133                 // LDS A-tile row stride (floats), conflict-avoiding
#define TT   4                   // pixel sub-tiles per block (64 pixels/block)

typedef __attribute__((ext_vector_type(2))) float v2f;
typedef __attribute__((ext_vector_type(8))) float v8f;

__device__ __forceinline__ int iclamp(int v, int lo, int hi) {
    return v < lo ? lo : (v > hi ? hi : v);
}

// ---------------------------------------------------------------------------
// 0) NCHW -> NHWC transpose of x
// ---------------------------------------------------------------------------
__global__ __launch_bounds__(256) void nchw_to_nhwc_kernel(
    const float* __restrict__ x, float* __restrict__ xn) {
    int idx = blockIdx.x * 256 + threadIdx.x;
    if (idx >= Bn * Cn * Hn * Wn) return;
    int px = idx % Wn; int r = idx / Wn;
    int py = r % Hn;   r /= Hn;
    int c  = r % Cn;   int bb = r / Cn;
    xn[(((size_t)bb * Hn + py) * Wn + px) * Cn + c] = x[idx];
}

// ---------------------------------------------------------------------------
// 1) weight packs:
//    w_def [OC][C][3][3] -> wdef_p [k][C/2][OC][2]  (K-pair contiguous per lane)
//    w_pw  [OC][C]       -> wpw_p  [C/2][OC][2]
// ---------------------------------------------------------------------------
__global__ __launch_bounds__(256) void wdef_pack_kernel(
    const float* __restrict__ w_def, float* __restrict__ wdef_p) {
    int idx = blockIdx.x * 256 + threadIdx.x;
    if (idx >= 9 * Cn * OCn) return;
    const int par = idx & 1; int r = idx >> 1;
    const int oc  = r % OCn; r /= OCn;
    const int ic2 = r % (Cn / 2); const int k = r / (Cn / 2);
    const int ic  = 2 * ic2 + par;
    wdef_p[idx] = w_def[((size_t)oc * Cn + ic) * 9 + k];
}

__global__ __launch_bounds__(256) void wpw_pack_kernel(
    const float* __restrict__ w_pw, float* __restrict__ wpw_p) {
    int idx = blockIdx.x * 256 + threadIdx.x;
    if (idx >= Cn * OCn) return;
    const int par = idx & 1; int r = idx >> 1;
    const int oc  = r % OCn; const int ic2 = r / OCn;
    const int ic  = 2 * ic2 + par;
    wpw_p[idx] = w_pw[(size_t)oc * Cn + ic];
}

// ---------------------------------------------------------------------------
// 2) offset conv: 3x3, C=128 -> 18, NHWC in, [pix][18] out. One block per (b,y).
// ---------------------------------------------------------------------------
__global__ __launch_bounds__(96) void offset_conv_kernel(
    const float* __restrict__ xn, const float* __restrict__ w_off,
    const float* __restrict__ b_off, float* __restrict__ off) {
    __shared__ float sw[OFFn * Cn];
    const int tid = threadIdx.x;
    const int by  = blockIdx.x;
    const int bb  = by / Hn, py = by % Hn;
    const int px  = tid;

    float acc[OFFn];
    for (int o = 0; o < OFFn; ++o) acc[o] = b_off[o];

    for (int k = 0; k < 9; ++k) {
        const int ky = k / 3, kx = k % 3;
        __syncthreads();
        for (int i = tid; i < OFFn * Cn; i += 96)
            sw[i] = w_off[(size_t)(i >> 7) * (Cn * 9) + (i & 127) * 9 + k];
        __syncthreads();
        const int yy = py - 1 + ky, xx = px - 1 + kx;
        if (yy >= 0 && yy < Hn && xx >= 0 && xx < Wn) {
            const float* xp = xn + (((size_t)bb * Hn + yy) * Wn + xx) * Cn;
            for (int ic = 0; ic < Cn; ++ic) {
                const float v = xp[ic];
                #pragma unroll
                for (int o = 0; o < OFFn; ++o) acc[o] += v * sw[o * Cn + ic];
            }
        }
    }
    const size_t pix = (size_t)(bb * Hn + py) * Wn + px;
    for (int o = 0; o < OFFn; ++o) off[pix * OFFn + o] = acc[o];
}

// ---------------------------------------------------------------------------
// 3) deformable conv (WMMA f32 16x16x4) + fused channel LayerNorm -> d_out NCHW
//    block = 256 threads = 8 wave32; tile = 64 pixels x 128 out-channels
//    (4 sub-tiles share each B fragment -> 4x less weight re-fetch)
// ---------------------------------------------------------------------------
__global__ __launch_bounds__(256) void deform_ln_kernel(
    const float* __restrict__ xn,      // [pix][C]
    const float* __restrict__ off,     // [pix][18]
    const float* __restrict__ wdef_p,  // [k][C/2][OC][2]
    const float* __restrict__ b_def, const float* __restrict__ ln_w,
    const float* __restrict__ ln_b, float* __restrict__ out) {
    __shared__ float Atile[16 * TT * AST];
    __shared__ float s_mean[16 * TT], s_rstd[16 * TT];

    const int tid  = threadIdx.x;
    const int lane = tid & 31;
    const int wave = tid >> 5;
    const int p0   = blockIdx.x * (16 * TT);

    // sampling role: 16 pixel-groups x 16 channel-threads, x TT sub-tiles
    const int grp = tid >> 4;
    const int lng = tid & 15;

    int pbb[TT], ppy[TT], ppx[TT];
    #pragma unroll
    for (int st = 0; st < TT; ++st) {
        const int pix = p0 + st * 16 + grp;
        pbb[st] = pix / HWn;
        const int rem = pix % HWn;
        ppy[st] = rem / Wn;
        ppx[st] = rem % Wn;
    }

    v8f acc[TT];
    #pragma unroll
    for (int st = 0; st < TT; ++st)
        acc[st] = (v8f){0.f, 0.f, 0.f, 0.f, 0.f, 0.f, 0.f, 0.f};

    const int n     = wave * 16 + (lane & 15);  // out channel for this lane
    const int khalf = (lane >> 4) * 2;          // K sub-block per ISA layout
    const int am    = lane & 15;                // A row within sub-tile

    for (int k = 0; k < 9; ++k) {
        const int ky = k / 3, kx = k % 3;
        __syncthreads();   // previous iteration's WMMA reads are done
        // ---- build bilinear-sampled A [64 pixels][128 ch] in LDS ----
        #pragma unroll
        for (int st = 0; st < TT; ++st) {
            const int pix = p0 + st * 16 + grp;
            const float dy = off[(size_t)pix * OFFn + 2 * k];
            const float dx = off[(size_t)pix * OFFn + 2 * k + 1];
            const float ys = (float)(ppy[st] - 1 + ky) + dy;
            const float xs = (float)(ppx[st] - 1 + kx) + dx;
            const float y0f = floorf(ys), x0f = floorf(xs);
            const int   y0  = (int)y0f,   x0  = (int)x0f;
            const float wy1 = ys - y0f, wx1 = xs - x0f;
            const float wy0 = 1.f - wy1, wx0 = 1.f - wx1;
            const bool vy0 = (y0 >= 0) && (y0 < Hn);
            const bool vy1 = (y0 + 1 >= 0) && (y0 + 1 < Hn);
            const bool vx0 = (x0 >= 0) && (x0 < Wn);
            const bool vx1 = (x0 + 1 >= 0) && (x0 + 1 < Wn);
            const float m00 = (vy0 && vx0) ? wy0 * wx0 : 0.f;
            const float m01 = (vy0 && vx1) ? wy0 * wx1 : 0.f;
            const float m10 = (vy1 && vx0) ? wy1 * wx0 : 0.f;
            const float m11 = (vy1 && vx1) ? wy1 * wx1 : 0.f;
            const int yc0 = iclamp(y0, 0, Hn - 1), yc1 = iclamp(y0 + 1, 0, Hn - 1);
            const int xc0 = iclamp(x0, 0, Wn - 1), xc1 = iclamp(x0 + 1, 0, Wn - 1);
            const size_t base = (size_t)pbb[st] * HWn;
            const float* p00 = xn + (base + (size_t)yc0 * Wn + xc0) * Cn;
            const float* p01 = xn + (base + (size_t)yc0 * Wn + xc1) * Cn;
            const float* p10 = xn + (base + (size_t)yc1 * Wn + xc0) * Cn;
            const float* p11 = xn + (base + (size_t)yc1 * Wn + xc1) * Cn;
            #pragma unroll
            for (int j = 0; j < 8; ++j) {
                const int ch = lng + 16 * j;
                Atile[(st * 16 + grp) * AST + ch] =
                    m00 * p00[ch] + m01 * p01[ch] + m10 * p10[ch] + m11 * p11[ch];
            }
        }
        __syncthreads();

        // ---- K = 128 dot: one coalesced b64 B-load feeds 4 WMMAs ----
        const v2f* wk = (const v2f*)(wdef_p + (size_t)k * (Cn / 2) * OCn * 2);
        for (int kk = 0; kk < Cn; kk += 4) {
            const int ic0 = kk + khalf;                       // even
            const v2f b = wk[(size_t)(ic0 >> 1) * OCn + n];   // (K, K+1) pair
            #pragma unroll
            for (int st = 0; st < TT; ++st) {
                v2f a;
                a.x = Atile[(st * 16 + am) * AST + ic0];
                a.y = Atile[(st * 16 + am) * AST + ic0 + 1];
                acc[st] = __builtin_amdgcn_wmma_f32_16x16x4_f32(
                    false, a, false, b, (short)0, acc[st], false, false);
            }
        }
    }

    // ---- epilogue: bias, LayerNorm over channels, NCHW store ----
    __syncthreads();
    const float bias = b_def[n];
    #pragma unroll
    for (int st = 0; st < TT; ++st) {
        #pragma unroll
        for (int r = 0; r < 8; ++r) {
            const int prow = st * 16 + r + 8 * (lane >> 4);  // C/D: M = r (+8 upper)
            Atile[prow * AST + n] = acc[st][r] + bias;
        }
    }
    __syncthreads();
    if (tid < 16 * TT) {
        float s = 0.f, sq = 0.f;
        for (int c = 0; c < Cn; ++c) {
            const float v = Atile[tid * AST + c];
            s += v; sq += v * v;
        }
        const float mean = s * (1.f / Cn);
        const float var  = sq * (1.f / Cn) - mean * mean;
        s_mean[tid] = mean;
        s_rstd[tid] = rsqrtf(var + 1e-6f);
    }
    __syncthreads();
    #pragma unroll
    for (int st = 0; st < TT; ++st) {
        const int prow = st * 16 + grp;
        const float mean = s_mean[prow], rstd = s_rstd[prow];
        #pragma unroll
        for (int j = 0; j < 8; ++j) {
            const int ch = lng + 16 * j;
            const float v =
                (Atile[prow * AST + ch] - mean) * rstd * ln_w[ch] + ln_b[ch];
            out[(((size_t)pbb[st] * OCn + ch) * Hn + ppy[st]) * Wn + ppx[st]] = v;
        }
    }
}

// ---------------------------------------------------------------------------
// 4) depthwise 3x3 + residual -> t (NHWC). One block per (b,c,y), thread per x.
// ---------------------------------------------------------------------------
__global__ __launch_bounds__(96) void dw_add_kernel(
    const float* __restrict__ x, const float* __restrict__ w_dw,
    const float* __restrict__ b_dw, float* __restrict__ t) {
    int bcy = blockIdx.x;
    const int py = bcy % Hn; int r = bcy / Hn;
    const int c  = r % Cn;   const int bb = r / Cn;
    const int px = threadIdx.x;
    const float* xp = x + ((size_t)(bb * Cn + c) * Hn) * Wn;
    float acc = b_dw[c];
    #pragma unroll
    for (int ky = 0; ky < 3; ++ky) {
        const int yy = py - 1 + ky;
        if (yy < 0 || yy >= Hn) continue;
        #pragma unroll
        for (int kx = 0; kx < 3; ++kx) {
            const int xx = px - 1 + kx;
            if (xx < 0 || xx >= Wn) continue;
            acc += w_dw[c * 9 + ky * 3 + kx] * xp[(size_t)yy * Wn + xx];
        }
    }
    acc += xp[(size_t)py * Wn + px];   // residual x
    t[(((size_t)bb * Hn + py) * Wn + px) * Cn + c] = acc;
}

// ---------------------------------------------------------------------------
// 5) pointwise 1x1 GEMM (WMMA f32 16x16x4) + bias + SiLU, accumulate into d_out
//    64 pixels x 128 oc per block; B fragment shared across 4 sub-tiles
// ---------------------------------------------------------------------------
__global__ __launch_bounds__(256) void pw_silu_add_kernel(
    const float* __restrict__ t,      // [pix][C]
    const float* __restrict__ wpw_p,  // [C/2][OC][2]
    const float* __restrict__ b_pw, float* __restrict__ out) {
    __shared__ float Atile[16 * TT * AST];
    const int tid  = threadIdx.x;
    const int lane = tid & 31;
    const int wave = tid >> 5;
    const int p0   = blockIdx.x * (16 * TT);

    for (int i = tid; i < 16 * TT * Cn; i += 256)
        Atile[(i >> 7) * AST + (i & 127)] = t[(size_t)p0 * Cn + i];
    __syncthreads();

    v8f acc[TT];
    #pragma unroll
    for (int st = 0; st < TT; ++st)
        acc[st] = (v8f){0.f, 0.f, 0.f, 0.f, 0.f, 0.f, 0.f, 0.f};

    const int n     = wave * 16 + (lane & 15);
    const int khalf = (lane >> 4) * 2;
    const int am    = lane & 15;
    const v2f* wv   = (const v2f*)wpw_p;
    for (int kk = 0; kk < Cn; kk += 4) {
        const int ic0 = kk + khalf;
        const v2f b = wv[(size_t)(ic0 >> 1) * OCn + n];
        #pragma unroll
        for (int st = 0; st < TT; ++st) {
            v2f a;
            a.x = Atile[(st * 16 + am) * AST + ic0];
            a.y = Atile[(st * 16 + am) * AST + ic0 + 1];
            acc[st] = __builtin_amdgcn_wmma_f32_16x16x4_f32(
                false, a, false, b, (short)0, acc[st], false, false);
        }
    }

    const float bias = b_pw[n];
    #pragma unroll
    for (int st = 0; st < TT; ++st) {
        #pragma unroll
        for (int r = 0; r < 8; ++r) {
            const int prow = st * 16 + r + 8 * (lane >> 4);
            const int pix  = p0 + prow;
            const int bb = pix / HWn, rem = pix % HWn;
            const int py = rem / Wn, px = rem % Wn;
            const float s   = acc[st][r] + bias;
            const float sig = 1.f / (1.f + __expf(-s));
            out[(((size_t)bb * OCn + n) * Hn + py) * Wn + px] += s * sig;
        }
    }
}

// ---------------------------------------------------------------------------
extern "C" void kernel_launch(void* const* d_in, const int* in_sizes, int n_in,
                              void* d_out, int out_size, void* d_ws, size_t ws_size,
                              hipStream_t stream) {
    (void)in_sizes; (void)n_in; (void)out_size; (void)ws_size;
    const float* x     = (const float*)d_in[0];
    const float* w_off = (const float*)d_in[1];
    const float* b_off = (const float*)d_in[2];
    const float* w_def = (const float*)d_in[3];
    const float* b_def = (const float*)d_in[4];
    const float* ln_w  = (const float*)d_in[5];
    const float* ln_b  = (const float*)d_in[6];
    const float* w_dw  = (const float*)d_in[7];
    const float* b_dw  = (const float*)d_in[8];
    const float* w_pw  = (const float*)d_in[9];
    const float* b_pw  = (const float*)d_in[10];
    float* out = (float*)d_out;

    // workspace layout (floats)
    float* ws     = (float*)d_ws;
    float* xn     = ws;                               // NPIX*C  (reused later as t)
    float* offb   = xn + (size_t)NPIX * Cn;           // NPIX*18
    float* wdef_p = offb + (size_t)NPIX * OFFn;       // 9*C*OC
    float* wpw_p  = wdef_p + (size_t)9 * Cn * OCn;    // C*OC
    float* tbuf   = xn;                               // reuse after deform_ln

    const int nelem = Bn * Cn * Hn * Wn;
    nchw_to_nhwc_kernel<<<(nelem + 255) / 256, 256, 0, stream>>>(x, xn);
    wdef_pack_kernel<<<(9 * Cn * OCn + 255) / 256, 256, 0, stream>>>(w_def, wdef_p);
    wpw_pack_kernel<<<(Cn * OCn + 255) / 256, 256, 0, stream>>>(w_pw, wpw_p);
    offset_conv_kernel<<<Bn * Hn, 96, 0, stream>>>(xn, w_off, b_off, offb);
    deform_ln_kernel<<<NPIX / (16 * TT), 256, 0, stream>>>(xn, offb, wdef_p, b_def,
                                                           ln_w, ln_b, out);
    dw_add_kernel<<<Bn * Cn * Hn, 96, 0, stream>>>(x, w_dw, b_dw, tbuf);
    pw_silu_add_kernel<<<NPIX / (16 * TT), 256, 0, stream>>>(tbuf, wpw_p, b_pw, out);
}